// TensorProductConvLayer_29815662969335
// MI455X (gfx1250) — compile-verified
//
#include <hip/hip_runtime.h>
#include <math.h>

// ---------------- problem constants (match reference) ----------------
#define N_NODES 10000
#define N_EDGES 100000
#define MUL0 32
#define MUL1 8
#define HID  64
#define WNUM 1920          // 1024+256+256+64+256+64
#define EPT  32            // edges per block tile (100000/32 = 3125 blocks exactly)
#define NTILES (WNUM/16)   // 120 WMMA column tiles

// OFFS = [0, 1024, 1280, 1536, 1600, 1856, 1920]
#define OFF_W1 0
#define OFF_W2 1024
#define OFF_W3 1280
#define OFF_W4 1536
#define OFF_W5 1600
#define OFF_W6 1856

#define ALPHA_C     0.15811388300841896660f   // 1/sqrt(40)
#define INV_SQRT3_C 0.57735026918962576451f

typedef float v2f __attribute__((ext_vector_type(2)));
typedef float v8f __attribute__((ext_vector_type(8)));

// ---------------- kernel 0: zero the node accumulator ----------------
__global__ void tpc_zero_agg(float* __restrict__ agg) {
  int i = blockIdx.x * blockDim.x + threadIdx.x;
  if (i < N_NODES * 64) agg[i] = 0.0f;
}

// ---------------- kernel 1: fused MLP(WMMA) + tensor product + scatter ----------------
__global__ __launch_bounds__(256) void tpc_edge_kernel(
    const float* __restrict__ node_attr,   // [N_NODES, 56]
    const int*   __restrict__ edge_index,  // [2, N_EDGES] (row0=src, row1=dst)
    const float* __restrict__ edge_attr,   // [N_EDGES, 4]
    const float* __restrict__ edge_feat,   // [N_EDGES, 32]
    const float* __restrict__ W1,          // [32, 64]
    const float* __restrict__ b1,          // [64]
    const float* __restrict__ W2,          // [64, 1920]
    const float* __restrict__ b2,          // [1920]
    float* __restrict__ agg)               // [N_NODES, 64]
{
  __shared__ float lds_h[EPT * HID];       //   8 KB  relu(ef@W1+b1)
  __shared__ float lds_w[EPT * WNUM];      // 240 KB  per-edge weights (WMMA output)
  __shared__ float lds_x0[EPT * MUL0];     //   4 KB
  __shared__ float lds_x1[EPT * MUL1 * 3]; //   3 KB
  __shared__ float lds_dot[EPT * MUL1];    //   1 KB
  __shared__ float lds_s0[EPT];
  __shared__ float lds_s1[EPT * 3];
  __shared__ int   lds_src[EPT];
  __shared__ int   lds_dst[EPT];

  const int tid   = threadIdx.x;
  const int eb    = blockIdx.x * EPT;      // first edge of this tile (always full tile)
  const int wave  = tid >> 5;              // 0..7
  const int lane  = tid & 31;
  const int m     = lane & 15;             // row (A) / column (B,C,D) within 16-tile
  const int khalf = lane >> 4;             // 0: K pair {k0,k0+1}, 1: K pair {k0+2,k0+3}

  // ---- phase 0a: per-edge scalars + indices ----
  if (tid < EPT) {
    const int e = eb + tid;
    lds_src[tid]     = edge_index[e];
    lds_dst[tid]     = edge_index[N_EDGES + e];
    lds_s0[tid]      = edge_attr[e * 4 + 0];
    lds_s1[tid*3+0]  = edge_attr[e * 4 + 1];
    lds_s1[tid*3+1]  = edge_attr[e * 4 + 2];
    lds_s1[tid*3+2]  = edge_attr[e * 4 + 3];
  }
  __syncthreads();

  // ---- phase 0b: gather node features for dst nodes (32 x 56 = 7 full passes) ----
  for (int idx = tid; idx < EPT * 56; idx += 256) {
    const int el = idx / 56, c = idx % 56;
    const float v = node_attr[(size_t)lds_dst[el] * 56 + c];
    if (c < MUL0) lds_x0[el * MUL0 + c] = v;
    else          lds_x1[el * 24 + (c - MUL0)] = v;
  }

  // ---- phase 1: h[32x64] = relu(ef[32x32] @ W1[32x64] + b1) via WMMA f32 16x16x4 ----
  // 8 tile jobs (2 M-tiles x 4 N-tiles) == 8 waves: one job per wave, no divergence.
  {
    const int mt  = wave >> 2;             // 0..1
    const int nt  = wave & 3;              // 0..3
    const int col = nt * 16 + m;
    v8f acc = {};
    #pragma unroll
    for (int ks = 0; ks < 8; ++ks) {
      const int k = ks * 4 + khalf * 2;
      v2f a, b;
      const float* efp = &edge_feat[(size_t)(eb + mt * 16 + m) * 32 + k];
      a.x = efp[0];
      a.y = efp[1];
      b.x = W1[k * HID + col];
      b.y = W1[(k + 1) * HID + col];
      acc = __builtin_amdgcn_wmma_f32_16x16x4_f32(
          false, a, false, b, (short)0, acc, false, false);
    }
    const float bias = b1[col];
    #pragma unroll
    for (int v = 0; v < 8; ++v) {
      const int row = v + khalf * 8;       // D layout: VGPR v, lane-half selects row+8
      const float hv = acc[v] + bias;
      lds_h[(mt * 16 + row) * HID + col] = hv > 0.0f ? hv : 0.0f;
    }
  }
  __syncthreads();

  // ---- phase 1b: dot[e][u] = (x1[e,u,:] . s1[e,:]) / sqrt(3); 32*8 = 256 == blockDim ----
  {
    const int el = tid >> 3, u = tid & 7;
    lds_dot[el * MUL1 + u] =
        (lds_x1[el*24 + u*3 + 0] * lds_s1[el*3 + 0] +
         lds_x1[el*24 + u*3 + 1] * lds_s1[el*3 + 1] +
         lds_x1[el*24 + u*3 + 2] * lds_s1[el*3 + 2]) * INV_SQRT3_C;
  }

  // ---- phase 2: w[32 x 1920] = h[32 x 64] @ W2[64 x 1920] + b2 via WMMA f32 16x16x4 ----
  // Dual M-tiles share every B fragment: halves W2/L2 traffic vs one 16-edge tile.
  {
    // Preload A fragments for both M-tiles (reused across this wave's 15 column tiles).
    v2f afrag0[16], afrag1[16];
    #pragma unroll
    for (int ks = 0; ks < 16; ++ks) {
      const int k = ks * 4 + khalf * 2;
      afrag0[ks].x = lds_h[m * HID + k];
      afrag0[ks].y = lds_h[m * HID + k + 1];
      afrag1[ks].x = lds_h[(16 + m) * HID + k];
      afrag1[ks].y = lds_h[(16 + m) * HID + k + 1];
    }
    // 120 tiles over 8 waves -> exactly 15 each: uniform trip count, EXEC all-ones.
    for (int nt = wave; nt < NTILES; nt += 8) {
      const int col = nt * 16 + m;
      v8f acc0 = {};
      v8f acc1 = {};
      #pragma unroll
      for (int ks = 0; ks < 16; ++ks) {
        const int k = ks * 4 + khalf * 2;
        v2f b;
        b.x = W2[(size_t)k * WNUM + col];
        b.y = W2[(size_t)(k + 1) * WNUM + col];
        acc0 = __builtin_amdgcn_wmma_f32_16x16x4_f32(
            false, afrag0[ks], false, b, (short)0, acc0, false, false);
        acc1 = __builtin_amdgcn_wmma_f32_16x16x4_f32(
            false, afrag1[ks], false, b, (short)0, acc1, false, false);
      }
      const float bias = b2[col];
      #pragma unroll
      for (int v = 0; v < 8; ++v) {
        const int row = v + khalf * 8;
        lds_w[row * WNUM + col]        = acc0[v] + bias;
        lds_w[(16 + row) * WNUM + col] = acc1[v] + bias;
      }
    }
  }
  __syncthreads();

  // ---- phase 3: tensor-product contraction + atomic scatter into agg ----
  for (int idx = tid; idx < EPT * 64; idx += 256) {
    const int el = idx >> 6, t = idx & 63;
    const float* w = &lds_w[el * WNUM];
    float acc;
    if (t < 32) {               // out_scalars[t] = s0*(x0 . w1[:,t]) + (dot . w2[:,t])
      float a0 = 0.0f, a1 = 0.0f;
      #pragma unroll
      for (int u = 0; u < 32; ++u) a0 += lds_x0[el*32 + u] * w[OFF_W1 + u*32 + t];
      #pragma unroll
      for (int u = 0; u < 8;  ++u) a1 += lds_dot[el*8 + u] * w[OFF_W2 + u*32 + t];
      acc = a0 * lds_s0[el] + a1;
    } else if (t < 40) {        // gates[t-32]
      const int tw = t - 32;
      float a0 = 0.0f, a1 = 0.0f;
      #pragma unroll
      for (int u = 0; u < 32; ++u) a0 += lds_x0[el*32 + u] * w[OFF_W3 + u*8 + tw];
      #pragma unroll
      for (int u = 0; u < 8;  ++u) a1 += lds_dot[el*8 + u] * w[OFF_W4 + u*8 + tw];
      acc = a0 * lds_s0[el] + a1;
    } else {                    // vec[wv][i], flat index 40 + wv*3 + i
      const int r = t - 40, wv = r / 3, i = r % 3;
      float a0 = 0.0f, a1 = 0.0f;
      #pragma unroll
      for (int u = 0; u < 32; ++u) a0 += lds_x0[el*32 + u]       * w[OFF_W5 + u*8 + wv];
      #pragma unroll
      for (int u = 0; u < 8;  ++u) a1 += lds_x1[el*24 + u*3 + i] * w[OFF_W6 + u*8 + wv];
      acc = a0 * lds_s1[el*3 + i] + a1 * lds_s0[el];
    }
    atomicAdd(&agg[(size_t)lds_src[el] * 64 + t], acc * ALPHA_C);
  }
}

// ---------------- kernel 2: silu / sigmoid-gated output ----------------
__global__ void tpc_finalize(const float* __restrict__ agg, float* __restrict__ out) {
  const int n = blockIdx.x * blockDim.x + threadIdx.x;
  if (n >= N_NODES) return;
  const float* a = &agg[(size_t)n * 64];
  float* o = &out[(size_t)n * 56];
  #pragma unroll
  for (int c = 0; c < 32; ++c) {
    const float x = a[c];
    o[c] = x / (1.0f + __expf(-x));        // silu
  }
  #pragma unroll
  for (int wv = 0; wv < 8; ++wv) {
    const float g = 1.0f / (1.0f + __expf(-a[32 + wv]));  // sigmoid gate
    #pragma unroll
    for (int i = 0; i < 3; ++i)
      o[32 + wv*3 + i] = g * a[40 + wv*3 + i];
  }
}

// ---------------- host entry ----------------
extern "C" void kernel_launch(void* const* d_in, const int* in_sizes, int n_in,
                              void* d_out, int out_size, void* d_ws, size_t ws_size,
                              hipStream_t stream) {
  (void)in_sizes; (void)n_in; (void)out_size; (void)ws_size;
  const float* node_attr  = (const float*)d_in[0];
  const int*   edge_index = (const int*)  d_in[1];
  const float* edge_attr  = (const float*)d_in[2];
  const float* edge_feat  = (const float*)d_in[3];
  const float* W1         = (const float*)d_in[4];
  const float* b1         = (const float*)d_in[5];
  const float* W2         = (const float*)d_in[6];
  const float* b2         = (const float*)d_in[7];
  float* out = (float*)d_out;
  float* agg = (float*)d_ws;               // N_NODES*64 floats = 2.56 MB scratch

  tpc_zero_agg<<<(N_NODES * 64 + 255) / 256, 256, 0, stream>>>(agg);
  tpc_edge_kernel<<<N_EDGES / EPT, 256, 0, stream>>>(
      node_attr, edge_index, edge_attr, edge_feat, W1, b1, W2, b2, agg);
  tpc_finalize<<<(N_NODES + 255) / 256, 256, 0, stream>>>(agg, out);
}